// simplified_gcn_contrastive_model_47278999994910
// MI455X (gfx1250) — compile-verified
//
#include <hip/hip_runtime.h>
#include <stdint.h>

typedef __attribute__((ext_vector_type(2))) float v2f;
typedef __attribute__((ext_vector_type(8))) float v8f;

#define N_NODES  100000
#define N_EDGES  3200000
#define EDGE_DIM 64
#define ADDED    24
#define ADDED_P  32             // zero-padded N for WMMA tile 1
#define H_DIM    256
#define OUT_DIM  256
#define X_DIM    280            // H_DIM + ADDED

// ---- CDNA5 async-to-LDS path (builtin confirmed present by probe) ----------
#if defined(__HIP_DEVICE_COMPILE__) && defined(__gfx1250__) && \
    __has_builtin(__builtin_amdgcn_global_load_async_to_lds_b128) && \
    __has_builtin(__builtin_amdgcn_s_wait_asynccnt)
#define USE_ASYNC_LDS 1
#else
#define USE_ASYNC_LDS 0
#endif

#if USE_ASYNC_LDS
typedef int v4i __attribute__((vector_size(16)));   // matches builtin param type
typedef v4i __attribute__((address_space(1))) g_v4i;
typedef v4i __attribute__((address_space(3))) l_v4i;
// Generic LDS pointer: low 32 bits are the LDS offset (aperture scheme), and
// AS(3) pointers are 32-bit, so an integer round-trip yields the LDS address.
// AS-qualified args also implicitly convert to generic params if needed.
__device__ __forceinline__ l_v4i* to_lds(void* p) {
  return (l_v4i*)(uintptr_t)p;
}
__device__ __forceinline__ g_v4i* to_gbl(const void* p) {
  return (g_v4i*)(uintptr_t)p;
}
#endif

// ---------------------------------------------------------------------------
// Edge kernel: m = edge_feat @ W_msg + b_msg, scatter-sum into ah[dst].
// 256 threads = 8 waves; each block processes 128 edges (8 tiles of 16 rows).
// ---------------------------------------------------------------------------
#define E_ROW_STRIDE 68         // 64 data floats + 4 pad -> conflict-free A reads

__global__ __launch_bounds__(256) void edge_msg_scatter(
    const float* __restrict__ edge_feat,
    const float* __restrict__ W_msg,
    const float* __restrict__ b_msg,
    const int*   __restrict__ dst_idx,
    float*       __restrict__ ah)
{
  __shared__ float elds[128 * E_ROW_STRIDE];
  __shared__ float wlds[EDGE_DIM * ADDED_P];   // zero-padded cols 24..31
  __shared__ float blds[ADDED_P];              // zero-padded bias
  __shared__ int   ilds[128];                  // dst indices for this tile

  const int tid  = threadIdx.x;
  const int lane = tid & 31;
  const int w    = tid >> 5;          // wave id 0..7 -> 16-edge tile
  const int half = lane >> 4;         // selects K pair / upper 8 D rows
  const int n    = lane & 15;         // N column (and A-frag M row)

  const long long E0 = (long long)blockIdx.x * 128;

  // Stage 128x64 edge_feat tile first (async engine fills LDS while we do the
  // small W_msg/bias/index staging with the VALU path).
  {
    const float4* g4 = (const float4*)(edge_feat + E0 * EDGE_DIM);
    for (int i = tid; i < 128 * (EDGE_DIM / 4); i += 256) {
      int r = i >> 4, c4 = i & 15;
#if USE_ASYNC_LDS
      __builtin_amdgcn_global_load_async_to_lds_b128(
          to_gbl(g4 + i), to_lds(&elds[r * E_ROW_STRIDE + c4 * 4]), 0, 0);
#else
      *(float4*)&elds[r * E_ROW_STRIDE + c4 * 4] = g4[i];
#endif
    }
  }

  // Stage W_msg zero-padded to 64x32 (single pass, no intra-block hazard)
  for (int i = tid; i < EDGE_DIM * ADDED_P; i += 256) {
    int r = i >> 5, c = i & 31;
    wlds[i] = (c < ADDED) ? W_msg[r * ADDED + c] : 0.0f;
  }
  if (tid < ADDED_P) blds[tid] = (tid < ADDED) ? b_msg[tid] : 0.0f;
  if (tid < 128)     ilds[tid] = dst_idx[E0 + tid];

#if USE_ASYNC_LDS
  __builtin_amdgcn_s_wait_asynccnt(0);
#endif
  __syncthreads();

  // C init = per-edge bias (bias applied per edge BEFORE the scatter-sum)
  v8f c0, c1;
  {
    float b0  = blds[n];
    float b1v = blds[16 + n];          // zero for padded cols 24..31
    for (int i = 0; i < 8; ++i) { c0[i] = b0; c1[i] = b1v; }
  }

  // A frag layout: lanes 0-15 -> M=n, K=4k+{0,1}; lanes 16-31 -> K=4k+{2,3}
  const int ldsrow = (w * 16 + n) * E_ROW_STRIDE;
#pragma unroll
  for (int k = 0; k < EDGE_DIM / 4; ++k) {
    const int kk = 4 * k;
    v2f a = *(const v2f*)&elds[ldsrow + kk + 2 * half];
    const int kr = kk + 2 * half;
    v2f bA, bB;                        // all loads unconditional (padded LDS)
    bA[0] = wlds[(kr + 0) * ADDED_P + n];
    bA[1] = wlds[(kr + 1) * ADDED_P + n];
    bB[0] = wlds[(kr + 0) * ADDED_P + 16 + n];
    bB[1] = wlds[(kr + 1) * ADDED_P + 16 + n];
    c0 = __builtin_amdgcn_wmma_f32_16x16x4_f32(false, a, false, bA, (short)0, c0, false, false);
    c1 = __builtin_amdgcn_wmma_f32_16x16x4_f32(false, a, false, bB, (short)0, c1, false, false);
  }

  // D layout: VGPR i holds row M = i + 8*half, col N = n (tile0) / 16+n (tile1).
  // ah is 9.6 MB -> L2-resident; unsafeAtomicAdd -> native global_atomic_add_f32.
#pragma unroll
  for (int i = 0; i < 8; ++i) {
    const int m = i + 8 * half;
    const int d = ilds[w * 16 + m];              // LDS broadcast
    float* dstp = ah + (long long)d * ADDED;
    unsafeAtomicAdd(dstp + n, c0[i]);
    if (n < 8) unsafeAtomicAdd(dstp + 16 + n, c1[i]);
  }
}

// ---------------------------------------------------------------------------
// Node kernel: out = relu([h | ah*norm] @ W1 + b1)
// One 16-node row tile per block; 8 waves x 2 col-tiles = 256 output cols.
// ---------------------------------------------------------------------------
#define X_STRIDE 292            // 280 data + pad; 292%64=36 -> 16 rows, 16 banks

__global__ __launch_bounds__(256) void node_update(
    const float* __restrict__ h,
    const float* __restrict__ norm,
    const float* __restrict__ ah,
    const float* __restrict__ W1,
    const float* __restrict__ b1,
    float*       __restrict__ out)
{
  __shared__ float xlds[16 * X_STRIDE];

  const int tid  = threadIdx.x;
  const int lane = tid & 31;
  const int w    = tid >> 5;
  const int half = lane >> 4;
  const int n    = lane & 15;
  const long long base = (long long)blockIdx.x * 16;

  // Stage h rows (16x256) — async DMA into LDS
  {
    const float4* g4 = (const float4*)(h + base * H_DIM);
    for (int i = tid; i < 16 * (H_DIM / 4); i += 256) {
      int r = i >> 6, c4 = i & 63;
#if USE_ASYNC_LDS
      __builtin_amdgcn_global_load_async_to_lds_b128(
          to_gbl(g4 + i), to_lds(&xlds[r * X_STRIDE + c4 * 4]), 0, 0);
#else
      *(float4*)&xlds[r * X_STRIDE + c4 * 4] = g4[i];
#endif
    }
  }
  // Stage ah * norm (16x24) into columns 256..279 (overlaps with async fill)
  for (int i = tid; i < 16 * ADDED; i += 256) {
    int r = i / ADDED, cc = i - r * ADDED;
    long long node = base + r;
    xlds[r * X_STRIDE + H_DIM + cc] = ah[node * ADDED + cc] * norm[node];
  }
#if USE_ASYNC_LDS
  __builtin_amdgcn_s_wait_asynccnt(0);
#endif
  __syncthreads();

  const int arow = n * X_STRIDE;
  for (int t = 0; t < 2; ++t) {
    const int col0 = (w + 8 * t) * 16;
    v8f c;
    { float bv = b1[col0 + n]; for (int i = 0; i < 8; ++i) c[i] = bv; }

#pragma unroll 5
    for (int k = 0; k < X_DIM / 4; ++k) {       // 70 K-steps of 4
      const int kk = 4 * k;
      v2f a = *(const v2f*)&xlds[arow + kk + 2 * half];
      const int kr = kk + 2 * half;
      v2f b;
      b[0] = W1[(long long)(kr + 0) * OUT_DIM + col0 + n];   // W1 is L2-hot (287 KB)
      b[1] = W1[(long long)(kr + 1) * OUT_DIM + col0 + n];
      c = __builtin_amdgcn_wmma_f32_16x16x4_f32(false, a, false, b, (short)0, c, false, false);
    }

#pragma unroll
    for (int i = 0; i < 8; ++i) {
      const int m = i + 8 * half;
      float v = c[i] > 0.0f ? c[i] : 0.0f;      // fused ReLU
      out[(base + m) * (long long)OUT_DIM + col0 + n] = v;
    }
  }
}

// ---------------------------------------------------------------------------
extern "C" void kernel_launch(void* const* d_in, const int* in_sizes, int n_in,
                              void* d_out, int out_size, void* d_ws, size_t ws_size,
                              hipStream_t stream) {
  const float* h         = (const float*)d_in[0];
  const float* edge_feat = (const float*)d_in[1];
  const float* norm      = (const float*)d_in[2];
  const float* W_msg     = (const float*)d_in[3];
  const float* b_msg     = (const float*)d_in[4];
  const float* W1        = (const float*)d_in[5];
  const float* b1        = (const float*)d_in[6];
  const int*   dst_idx   = (const int*)d_in[7];
  float*       out       = (float*)d_out;
  float*       ah        = (float*)d_ws;       // [N_NODES, 24] accumulator

  (void)hipMemsetAsync(ah, 0, sizeof(float) * (size_t)N_NODES * ADDED, stream);

  edge_msg_scatter<<<N_EDGES / 128, 256, 0, stream>>>(edge_feat, W_msg, b_msg,
                                                      dst_idx, ah);
  node_update<<<N_NODES / 16, 256, 0, stream>>>(h, norm, ah, W1, b1, out);
}